// StochasticNeighborLoss_39573828665540
// MI455X (gfx1250) — compile-verified
//
#include <hip/hip_runtime.h>
#include <hip/hip_bf16.h>
#include <stdint.h>

// Problem constants (from reference): N=50000, K=32, D=128
#define SNL_N 50000
#define SNL_K 32
#define SNL_D 128
#define ROWS_PER_BLOCK 2            // 2 waves of 32 -> 64 threads/block
#define NBLOCKS (SNL_N / ROWS_PER_BLOCK)   // 25000 partials
#define STRF 132                    // floats per staged anchor row (528B: pad -> 2-cycle ds_load_b128, no extra conflicts)
#define RED_PAD 25024               // 391 * 64, padded partial count for the WMMA reduction
#define RED_CHUNKS (RED_PAD / 64)   // 391

typedef float v2f __attribute__((ext_vector_type(2)));
typedef float v8f __attribute__((ext_vector_type(8)));

// ---- CDNA5 async global->LDS copy (16B per lane, one instruction = 512B/wave) ----
// Inline asm form is portable across ROCm7.2 clang-22 and amdgpu-toolchain clang-23.
__device__ __forceinline__ void async_copy_b128(const float* g, const float* lds_generic) {
    unsigned ldsOff = (unsigned)(size_t)(const void*)lds_generic;   // low 32b of generic LDS ptr = LDS byte offset
    unsigned long long ga = (unsigned long long)(size_t)(const void*)g;
    asm volatile("global_load_async_to_lds_b128 %0, %1, off"
                 :: "v"(ldsOff), "v"(ga)
                 : "memory");
}
__device__ __forceinline__ void wait_async0() {
    asm volatile("s_wait_asynccnt 0x0" ::: "memory");
}

// ------------------------------------------------------------------
// Kernel 1: one wave32 per row. lane == anchor index k.
//  - stage yi (512B) + 32 anchor rows (512B each) into LDS via async-to-LDS
//    in anchor-major order (coalesced 4x128B requests per instruction)
//  - d2 accumulation in f32 from LDS, softmax via wave32 shfl_xor butterflies
//  - q written coalesced; per-block loss partial -> d_ws (deterministic tree)
// ------------------------------------------------------------------
__global__ void __launch_bounds__(ROWS_PER_BLOCK * 32)
snl_main_kernel(const float* __restrict__ emb,
                const float* __restrict__ P,
                const int*   __restrict__ aidx,
                float*       __restrict__ out,
                float*       __restrict__ ws)
{
    __shared__ __align__(16) float anc[ROWS_PER_BLOCK][SNL_K * STRF];
    __shared__ __align__(16) float yib[ROWS_PER_BLOCK][SNL_D];
    __shared__ float lacc[ROWS_PER_BLOCK];

    const int lane = threadIdx.x & 31;
    const int w    = threadIdx.x >> 5;
    const int row  = blockIdx.x * ROWS_PER_BLOCK + w;

    // coalesced per-lane row data
    const int   my_idx = aidx[(size_t)row * SNL_K + lane];
    const float pk     = P[(size_t)row * SNL_K + lane];

    // stage yi: 32 lanes x 16B = 512B in one async op
    async_copy_b128(emb + (size_t)row * SNL_D + lane * 4, &yib[w][lane * 4]);

    // stage the 32 anchor rows, anchor-major (each instr: one anchor's 512B, coalesced)
#pragma unroll
    for (int a = 0; a < SNL_K; ++a) {
        int ia = __shfl(my_idx, a, 32);   // wave-uniform anchor id (v_readlane)
        async_copy_b128(emb + (size_t)ia * SNL_D + lane * 4,
                        &anc[w][a * STRF + lane * 4]);
    }
    wait_async0();   // per-wave ASYNCcnt, private buffers -> no barrier needed

    // d2 for this lane's anchor (exact f32)
    float acc = 0.f;
#pragma unroll
    for (int t = 0; t < SNL_D / 4; ++t) {
        float4 yi4 = *(const float4*)&yib[w][t * 4];
        float4 b4  = *(const float4*)&anc[w][lane * STRF + t * 4];
        float e0 = yi4.x - b4.x, e1 = yi4.y - b4.y;
        float e2 = yi4.z - b4.z, e3 = yi4.w - b4.w;
        acc = fmaf(e0, e0, acc); acc = fmaf(e1, e1, acc);
        acc = fmaf(e2, e2, acc); acc = fmaf(e3, e3, acc);
    }

    // log-softmax over the 32 lanes (z = -d2)
    float z = -acc;
    float m = z;
#pragma unroll
    for (int off = 16; off; off >>= 1) m = fmaxf(m, __shfl_xor(m, off, 32));
    float e = __expf(z - m);
    float s = e;
#pragma unroll
    for (int off = 16; off; off >>= 1) s += __shfl_xor(s, off, 32);
    float logq = (z - m) - __logf(s);

    // q output (coalesced). d_out[0] is the loss slot.
    out[1 + (size_t)row * SNL_K + lane] = e / s;

    // KL term, wave-reduced
    float term = pk * (__logf(pk + 1e-12f) - logq);
#pragma unroll
    for (int off = 16; off; off >>= 1) term += __shfl_xor(term, off, 32);
    if (lane == 0) lacc[w] = term;
    __syncthreads();
    if (threadIdx.x == 0) ws[blockIdx.x] = lacc[0] + lacc[1];
}

// ------------------------------------------------------------------
// Kernel 2: deterministic reduction of 25000 block partials.
// Uses v_wmma_f32_16x16x4_f32 with B = all-ones (layout-invariant):
// D[m][n] = sum_k A[m][k] -> exact f32 row sums, accumulated in C.
// ------------------------------------------------------------------
__global__ void __launch_bounds__(256)
snl_reduce_kernel(float* __restrict__ ws, float* __restrict__ out)
{
    const int tid  = threadIdx.x;
    const int lane = tid & 31;
    const int w    = tid >> 5;
    __shared__ float sacc[8];

    // zero the pad tail every call (ws is not re-initialized by the harness)
    for (int i = NBLOCKS + tid; i < RED_PAD; i += 256) ws[i] = 0.f;
    __syncthreads();

#if __has_builtin(__builtin_amdgcn_wmma_f32_16x16x4_f32)
    v8f acc = {};
    v2f ones; ones[0] = 1.f; ones[1] = 1.f;
    for (int c = w; c < RED_CHUNKS; c += 8) {          // wave-uniform loop, EXEC all ones
        v2f a;
        a[0] = ws[c * 64 + lane];
        a[1] = ws[c * 64 + 32 + lane];
        acc = __builtin_amdgcn_wmma_f32_16x16x4_f32(false, a, false, ones,
                                                    (short)0, acc, false, false);
    }
    float t = acc[0] + acc[1] + acc[2] + acc[3] + acc[4] + acc[5] + acc[6] + acc[7];
    t += __shfl_xor(t, 16, 32);          // lanes<16 hold M=0..7 sums, lanes>=16 hold M=8..15
#else
    float t = 0.f;
    for (int c = w; c < RED_CHUNKS; c += 8)
        t += ws[c * 64 + lane] + ws[c * 64 + 32 + lane];
#pragma unroll
    for (int off = 16; off; off >>= 1) t += __shfl_xor(t, off, 32);
#endif

    if (lane == 0) sacc[w] = t;
    __syncthreads();
    if (tid == 0) {
        float s = 0.f;
        for (int j = 0; j < 8; ++j) s += sacc[j];
        out[0] = s / (float)SNL_N;
    }
}

extern "C" void kernel_launch(void* const* d_in, const int* in_sizes, int n_in,
                              void* d_out, int out_size, void* d_ws, size_t ws_size,
                              hipStream_t stream)
{
    const float* emb = (const float*)d_in[0];   // output_embedding [N, D] f32
    const float* P   = (const float*)d_in[1];   // input_similarity [N, K] f32
    const int*   idx = (const int*)d_in[2];     // anchor_idx [N, K] int32
    float* out = (float*)d_out;                 // [0]=loss, [1..]=q (N*K)
    float* ws  = (float*)d_ws;                  // >= RED_PAD floats

    snl_main_kernel<<<dim3(NBLOCKS), dim3(ROWS_PER_BLOCK * 32), 0, stream>>>(emb, P, idx, out, ws);
    snl_reduce_kernel<<<dim3(1), dim3(256), 0, stream>>>(ws, out);
}